// SelectiveGatingMechanism_65532611002879
// MI455X (gfx1250) — compile-verified
//
#include <hip/hip_runtime.h>
#include <stdint.h>

typedef __attribute__((ext_vector_type(2))) float v2f;
typedef __attribute__((ext_vector_type(8))) float v8f;
typedef __attribute__((ext_vector_type(4))) int   v4i;

#define B_DIM 16384
#define IN_DIM 1024
#define H_DIM 2048

#ifndef __has_builtin
#define __has_builtin(x) 0
#endif

#if __has_builtin(__builtin_amdgcn_global_load_async_to_lds_b128)
#define HAVE_ASYNC_LDS 1
#else
#define HAVE_ASYNC_LDS 0
#endif

typedef __attribute__((address_space(1))) v4i gv4i;   // global int4
typedef __attribute__((address_space(3))) v4i lv4i;   // LDS int4

__device__ __forceinline__ float sigmoidf_fast(float x) {
    return 1.0f / (1.0f + __expf(-x));
}

// D = A(16x4,f32) * B(4x16,f32) + C(16x16,f32): full fp32 on the matrix pipe.
__device__ __forceinline__ v8f wmma_f32(v2f a, v2f b, v8f c) {
    return __builtin_amdgcn_wmma_f32_16x16x4_f32(
        /*neg_a=*/false, a, /*neg_b=*/false, b,
        /*c_mod=*/(short)0, c, /*reuse_a=*/false, /*reuse_b=*/false);
}

// Stage a 64-row x 32-K fp32 chunk into an LDS buffer.
// Rows padded to 36 floats (144 B) so every float4-slot is 16B-aligned ->
// legal target for GLOBAL_LOAD_ASYNC_TO_LDS_B128 (ASYNCcnt DMA, no VGPR hop).
__device__ __forceinline__ void stage(float (*buf)[36], const float* __restrict__ src,
                                      int rowBase, int k0, int ld) {
    const int tid = threadIdx.x;
    const int r  = tid >> 3;         // 0..31
    const int kk = (tid & 7) << 2;   // 0,4,...,28
#pragma unroll
    for (int h = 0; h < 2; ++h) {
        const float* g = src + (size_t)(rowBase + r + h * 32) * ld + (k0 + kk);
        float* l = &buf[r + h * 32][kk];
#if HAVE_ASYNC_LDS
        __builtin_amdgcn_global_load_async_to_lds_b128(
            (gv4i*)(float*)g, (lv4i*)l, /*offset=*/0, /*cpol=*/0);
#else
        const float4 v = *(const float4*)g;
        l[0] = v.x; l[1] = v.y; l[2] = v.z; l[3] = v.w;
#endif
    }
}

__device__ __forceinline__ void wait_async0() {
#if HAVE_ASYNC_LDS
#if __has_builtin(__builtin_amdgcn_s_wait_asynccnt)
    __builtin_amdgcn_s_wait_asynccnt(0);
#else
    asm volatile("s_wait_asynccnt 0x0" ::: "memory");
#endif
#endif
}

// ---------------------------------------------------------------------------
// Double-buffered K-phase: 4 gate accumulators sharing one staged A tile.
// A = src[B, H]; weight rows offset by wRowOff (0 = proj-half, H = hidden-half).
// ---------------------------------------------------------------------------
__device__ __forceinline__ void phase4(float (*lds)[64][36],
                                       const float* __restrict__ A,
                                       const float* __restrict__ Wf,
                                       const float* __restrict__ Wi,
                                       const float* __restrict__ Wc,
                                       const float* __restrict__ Wo,
                                       int rowBase, int wRowOff,
                                       int col, int lrow, int koff,
                                       v8f& af, v8f& ai, v8f& ac, v8f& ao) {
    stage(lds[0], A, rowBase, 0, H_DIM);
    wait_async0();
    __syncthreads();
    for (int k0 = 0; k0 < H_DIM; k0 += 32) {
        const int cur = (k0 >> 5) & 1;
        if (k0 + 32 < H_DIM) stage(lds[cur ^ 1], A, rowBase, k0 + 32, H_DIM);
#pragma unroll
        for (int kk = 0; kk < 32; kk += 4) {
            v2f a;
            a.x = lds[cur][lrow][kk + koff];
            a.y = lds[cur][lrow][kk + koff + 1];
            const size_t gk = (size_t)(wRowOff + k0 + kk + koff);
            v2f b0, b1, b2, b3;
            b0.x = Wf[gk * H_DIM + col];  b0.y = Wf[(gk + 1) * H_DIM + col];
            b1.x = Wi[gk * H_DIM + col];  b1.y = Wi[(gk + 1) * H_DIM + col];
            b2.x = Wc[gk * H_DIM + col];  b2.y = Wc[(gk + 1) * H_DIM + col];
            b3.x = Wo[gk * H_DIM + col];  b3.y = Wo[(gk + 1) * H_DIM + col];
            af = wmma_f32(a, b0, af);
            ai = wmma_f32(a, b1, ai);
            ac = wmma_f32(a, b2, ac);
            ao = wmma_f32(a, b3, ao);
        }
        wait_async0();
        __syncthreads();
    }
}

// Double-buffered K-phase, single accumulator (proj / err-attention GEMM).
__device__ __forceinline__ void phase1(float (*lds)[64][36],
                                       const float* __restrict__ A,
                                       const float* __restrict__ W,
                                       int rowBase, int kTotal, int ld,
                                       int col, int lrow, int koff,
                                       v8f& acc) {
    stage(lds[0], A, rowBase, 0, ld);
    wait_async0();
    __syncthreads();
    for (int k0 = 0; k0 < kTotal; k0 += 32) {
        const int cur = (k0 >> 5) & 1;
        if (k0 + 32 < kTotal) stage(lds[cur ^ 1], A, rowBase, k0 + 32, ld);
#pragma unroll
        for (int kk = 0; kk < 32; kk += 4) {
            v2f a;
            a.x = lds[cur][lrow][kk + koff];
            a.y = lds[cur][lrow][kk + koff + 1];
            const size_t gk = (size_t)(k0 + kk + koff);
            v2f b;
            b.x = W[gk * H_DIM + col];
            b.y = W[(gk + 1) * H_DIM + col];
            acc = wmma_f32(a, b, acc);
        }
        wait_async0();
        __syncthreads();
    }
}

// ---------------------------------------------------------------------------
// Kernel 1: proj = input[B,IN] @ W_proj[IN,H] + b_proj   (fp32 WMMA)
// 256 thr = 8 waves (4 row x 2 col); block tile 64 rows x 32 cols.
// ---------------------------------------------------------------------------
__global__ __launch_bounds__(256)
void proj_kernel(const float* __restrict__ X, const float* __restrict__ W,
                 const float* __restrict__ bias, float* __restrict__ Y) {
    __shared__ float lds[2][64][36];
    const int lane = threadIdx.x & 31;
    const int wave = threadIdx.x >> 5;
    const int wrow = wave & 3;
    const int wcol = wave >> 2;
    const int rowBase = blockIdx.y * 64;
    const int colBase = blockIdx.x * 32;

    const int col  = colBase + wcol * 16 + (lane & 15);
    const int lrow = wrow * 16 + (lane & 15);
    const int koff = (lane < 16) ? 0 : 2;   // ISA 16x4 f32 A-fragment layout

    v8f acc = {};
    phase1(lds, X, W, rowBase, IN_DIM, IN_DIM, col, lrow, koff, acc);

    const float bv = bias[col];
    const int rb = rowBase + wrow * 16 + ((lane >> 4) << 3);
#pragma unroll
    for (int v = 0; v < 8; ++v) {
        Y[(size_t)(rb + v) * H_DIM + col] = acc[v] + bv;
    }
}

// ---------------------------------------------------------------------------
// Kernel 2: fused gates + err-attention + cell update.
// 5 fp32 WMMA accumulators per wave; A tiles staged via async-to-LDS DMA,
// double buffered; weights read from L2-resident global.
// ---------------------------------------------------------------------------
__global__ __launch_bounds__(256)
void gates_kernel(const float* __restrict__ proj,
                  const float* __restrict__ hidden,
                  const float* __restrict__ cell_in,
                  const float* __restrict__ err_sig,
                  const float* __restrict__ unc,
                  const float* __restrict__ Wf, const float* __restrict__ bf,
                  const float* __restrict__ Wi, const float* __restrict__ bi,
                  const float* __restrict__ Wc, const float* __restrict__ bc,
                  const float* __restrict__ Wo, const float* __restrict__ bo,
                  const float* __restrict__ We, const float* __restrict__ be,
                  const float* __restrict__ ts,
                  float* __restrict__ out_hidden,
                  float* __restrict__ out_cell) {
    __shared__ float lds[2][64][36];
    const int lane = threadIdx.x & 31;
    const int wave = threadIdx.x >> 5;
    const int wrow = wave & 3;
    const int wcol = wave >> 2;
    const int rowBase = blockIdx.y * 64;
    const int colBase = blockIdx.x * 32;

    const int col  = colBase + wcol * 16 + (lane & 15);
    const int lrow = wrow * 16 + (lane & 15);
    const int koff = (lane < 16) ? 0 : 2;

    v8f af = {}, ai = {}, ac = {}, ao = {}, ae = {};

    // Phase 1: A = proj, weight rows [0, H)
    phase4(lds, proj,   Wf, Wi, Wc, Wo, rowBase, 0,     col, lrow, koff, af, ai, ac, ao);
    // Phase 2: A = hidden_state, weight rows [H, 2H)
    phase4(lds, hidden, Wf, Wi, Wc, Wo, rowBase, H_DIM, col, lrow, koff, af, ai, ac, ao);
    // Phase 3: A = error_signal vs W_err
    phase1(lds, err_sig, We, rowBase, H_DIM, H_DIM, col, lrow, koff, ae);

    // ---- Epilogue: gating + cell update ----
    const float tscale = ts[0];
    const float bfv = bf[col];
    const float biv = bi[col];
    const float bcv = bc[col];
    const float bov = bo[col];
    const float bev = be[col];
    const int rb = rowBase + wrow * 16 + ((lane >> 4) << 3);

#pragma unroll
    for (int v = 0; v < 8; ++v) {
        const size_t idx = (size_t)(rb + v) * H_DIM + col;
        const float fg   = sigmoidf_fast(af[v] + bfv);
        const float ig   = sigmoidf_fast(ai[v] + biv);
        const float cand = tanhf(ac[v] + bcv);
        const float og   = sigmoidf_fast(ao[v] + bov);
        const float ea   = sigmoidf_fast(ae[v] + bev);
        const float u  = unc[idx];
        const float cs = cell_in[idx];
        const float mod_i = ig * (1.0f + u);
        const float mod_f = fg * (1.0f - 0.1f * u);
        const float cell  = (mod_f * cs + mod_i * cand * ea) * tscale;
        const float hid   = og * tanhf(cell);
        out_hidden[idx] = hid;
        out_cell[idx]   = cell;
    }
}

extern "C" void kernel_launch(void* const* d_in, const int* in_sizes, int n_in,
                              void* d_out, int out_size, void* d_ws, size_t ws_size,
                              hipStream_t stream) {
    const float* input  = (const float*)d_in[0];
    const float* hidden = (const float*)d_in[1];
    const float* cell   = (const float*)d_in[2];
    const float* errs   = (const float*)d_in[3];
    const float* unc    = (const float*)d_in[4];
    const float* Wp     = (const float*)d_in[5];
    const float* bp     = (const float*)d_in[6];
    const float* Wf     = (const float*)d_in[7];
    const float* bf     = (const float*)d_in[8];
    const float* Wi     = (const float*)d_in[9];
    const float* bi     = (const float*)d_in[10];
    const float* Wc     = (const float*)d_in[11];
    const float* bc     = (const float*)d_in[12];
    const float* Wo     = (const float*)d_in[13];
    const float* bo     = (const float*)d_in[14];
    const float* We     = (const float*)d_in[15];
    const float* be     = (const float*)d_in[16];
    const float* ts     = (const float*)d_in[17];

    float* out_hidden = (float*)d_out;
    float* out_cell   = out_hidden + (size_t)B_DIM * H_DIM;
    float* proj       = (float*)d_ws;   // B*H fp32 = 128 MB scratch

    dim3 blk(256);
    dim3 grd(H_DIM / 32, B_DIM / 64);   // 64 x 256 blocks; tiles divide evenly

    proj_kernel<<<grd, blk, 0, stream>>>(input, Wp, bp, proj);
    gates_kernel<<<grd, blk, 0, stream>>>(proj, hidden, cell, errs, unc,
                                          Wf, bf, Wi, bi, Wc, bc, Wo, bo,
                                          We, be, ts, out_hidden, out_cell);
}